// HAO_42923903156378
// MI455X (gfx1250) — compile-verified
//
#include <hip/hip_runtime.h>
#include <math.h>

typedef __attribute__((ext_vector_type(2))) float v2f;
typedef __attribute__((ext_vector_type(4))) float v4f;
typedef __attribute__((ext_vector_type(8))) float v8f;

#define F_DIM 128
#define N_WIN 64
#define CDIM  8
#define D_DIM 22016          // CD*F + CD*N + F*F + N*N
#define NF    (N_WIN * F_DIM) // 8192
// vec layout: [t_f 512 | s_f 1024 | adj_t 4096 | adj_s 16384]
#define VEC_TF   0
#define VEC_SF   512
#define VEC_ADJT 1536
#define VEC_ADJS 5632
#define FLT_MAX_ 3.4028234663852886e38f

__device__ __forceinline__ float sig_(float x) { return 1.f / (1.f + __expf(-x)); }

// Branchless jnp.nan_to_num: NaN -> 0, +/-inf -> +/-FLT_MAX.
// Lowers to v_max/v_min/v_cmp_o/v_cndmask, no EXEC divergence in the WMMA loop.
__device__ __forceinline__ float nan0_(float v) {
  float c = fminf(fmaxf(v, -FLT_MAX_), FLT_MAX_); // clamps +/-inf, NaN -> -FLT_MAX (fixed next)
  return (v == v) ? c : 0.f;                      // ordered-compare select kills NaN
}

// ---------------------------------------------------------------------------
// Out[m][n] = act( sum_k A[m][k] * W[n][k] )   via V_WMMA_F32_16X16X4_F32.
// One wave per 16x16 output tile. A-matrix (16x4 f32) lane layout per ISA:
// lanes 0..15 hold M=0..15 with K={0,1}; lanes 16..31 hold M=0..15 with K={2,3}
// (each lane loads 2 consecutive floats -> global_load_b64). B = W^T mirrored:
// each lane reads 2 consecutive floats from W row (nBase+l) at the same K pair.
// C/D: VGPR r holds M=r (lanes 0-15) and M=8+r (lanes 16-31), N = lane&15.
// ACT: 0 = relu, 1 = sigmoid.  SANITIZE: nan_to_num on A (compile-time).
// ---------------------------------------------------------------------------
template <int ACT, int SANITIZE>
__global__ void wmma_linear_f32(const float* __restrict__ A, int lda,
                                const float* __restrict__ W, int ldw,
                                float* __restrict__ C, int ldc,
                                int nTilesX, int K) {
  const int lane = threadIdx.x & 31;
  const int wave = (blockIdx.x * blockDim.x + threadIdx.x) >> 5;
  const int mBase = (wave / nTilesX) * 16;
  const int nBase = (wave % nTilesX) * 16;
  const int half = lane >> 4;   // selects K pair {0,1} vs {2,3}
  const int l    = lane & 15;

  const float* __restrict__ arow = A + (size_t)(mBase + l) * lda + 2 * half;
  const float* __restrict__ wrow = W + (size_t)(nBase + l) * ldw + 2 * half;

  v8f acc = {0.f, 0.f, 0.f, 0.f, 0.f, 0.f, 0.f, 0.f};
#pragma unroll 4
  for (int k = 0; k < K; k += 4) {
    v2f a, b;
    a.x = arow[k]; a.y = arow[k + 1];
    b.x = wrow[k]; b.y = wrow[k + 1];
    if (SANITIZE) { a.x = nan0_(a.x); a.y = nan0_(a.y); }
    acc = __builtin_amdgcn_wmma_f32_16x16x4_f32(false, a, false, b,
                                                (short)0, acc, false, false);
  }

  float* crow = C + (size_t)(mBase + 8 * half) * ldc + nBase + l;
#pragma unroll
  for (int r = 0; r < 8; ++r) {
    float v = acc[r];
    v = ACT ? sig_(v) : fmaxf(v, 0.f);
    crow[(size_t)r * ldc] = v;
  }
}

// ---------------------------------------------------------------------------
// tmp_t[j][c] = sum_k xc[j][k]*Wt[c][k]   (64x8,  K=128)
// tmp_s[f][c] = sum_n xc[n][f]*Ws[c][n]   (128x8, K=64)
// ---------------------------------------------------------------------------
__global__ void proj_small(const float* __restrict__ xc,
                           const float* __restrict__ Wt,
                           const float* __restrict__ Ws,
                           float* __restrict__ tmp_t, float* __restrict__ tmp_s) {
  const int t = blockIdx.x * blockDim.x + threadIdx.x;
  if (t < 512) {
    const int j = t >> 3, c = t & 7;
    float s = 0.f;
    for (int k = 0; k < F_DIM; ++k) s = fmaf(xc[j * F_DIM + k], Wt[c * F_DIM + k], s);
    tmp_t[t] = s;
  } else if (t < 1536) {
    const int u = t - 512, f = u >> 3, c = u & 7;
    float s = 0.f;
    for (int n = 0; n < N_WIN; ++n) s = fmaf(xc[n * F_DIM + f], Ws[c * N_WIN + n], s);
    tmp_s[u] = s;
  }
}

// t_f[n][c] = relu(adj_t[n][:] . tmp_t[:][c]) -> vec[n*8+c]
// s_f[f][c] = relu(adj_s[f][:] . tmp_s[:][c]) -> vec[512 + f*8+c]
__global__ void agg_small(float* __restrict__ vec,
                          const float* __restrict__ tmp_t,
                          const float* __restrict__ tmp_s) {
  const int t = blockIdx.x * blockDim.x + threadIdx.x;
  const float* adjt = vec + VEC_ADJT;
  const float* adjs = vec + VEC_ADJS;
  if (t < 512) {
    const int n = t >> 3, c = t & 7;
    float s = 0.f;
    for (int j = 0; j < N_WIN; ++j) s = fmaf(adjt[n * N_WIN + j], tmp_t[j * 8 + c], s);
    vec[VEC_TF + t] = fmaxf(s, 0.f);
  } else if (t < 1536) {
    const int u = t - 512, f = u >> 3, c = u & 7;
    float s = 0.f;
    for (int j = 0; j < F_DIM; ++j) s = fmaf(adjs[f * F_DIM + j], tmp_s[j * 8 + c], s);
    vec[VEC_SF + u] = fmaxf(s, 0.f);
  }
}

// ---------------------------------------------------------------------------
// Dominant GEMV 1: h[row] = relu(Wout[row,:] . vec), rows = 22016.
// One wave per row; float4 coalesced, non-temporal weight stream (1.94 GB used
// once — keep it out of L2 so the 88 KB vec stays hot), global_prefetch ahead.
// Fused epilogue: sigmoid slices of h -> d_out (t_out, s_out).
// ---------------------------------------------------------------------------
__global__ void gemv_out(const float* __restrict__ W, const float* __restrict__ v,
                         float* __restrict__ h, float* __restrict__ dst) {
  const int lane = threadIdx.x & 31;
  const int row  = (blockIdx.x * blockDim.x + threadIdx.x) >> 5;
  const v4f* __restrict__ wr = (const v4f*)(W + (size_t)row * D_DIM);
  const v4f* __restrict__ vr = (const v4f*)v;
  float acc = 0.f;
  for (int i = lane; i < D_DIM / 4; i += 32) {
    __builtin_prefetch((const void*)(wr + i + 64), 0, 0);  // global_prefetch_b8 ahead
    v4f w  = __builtin_nontemporal_load(wr + i);
    v4f xv = vr[i];
    acc = fmaf(w.x, xv.x, acc);
    acc = fmaf(w.y, xv.y, acc);
    acc = fmaf(w.z, xv.z, acc);
    acc = fmaf(w.w, xv.w, acc);
  }
#pragma unroll
  for (int off = 16; off; off >>= 1) acc += __shfl_xor(acc, off, 32);
  if (lane == 0) {
    const float o = fmaxf(acc, 0.f);
    h[row] = o;
    if (row >= VEC_ADJT && row < VEC_ADJS)
      dst[NF + (row - VEC_ADJT)] = sig_(o);                    // t_out (4096)
    else if (row >= VEC_ADJS)
      dst[NF + 4096 + (row - VEC_ADJS)] = sig_(o);             // s_out (16384)
  }
}

// GEMV 2: dst[row] = relu(Wfcn[row,:] . h + b[row]), rows = 8192. + curvs tail.
__global__ void gemv_fcn(const float* __restrict__ W, const float* __restrict__ h,
                         const float* __restrict__ b, const float* __restrict__ curvs,
                         float* __restrict__ dst) {
  const int lane = threadIdx.x & 31;
  const int row  = (blockIdx.x * blockDim.x + threadIdx.x) >> 5;
  const v4f* __restrict__ wr = (const v4f*)(W + (size_t)row * D_DIM);
  const v4f* __restrict__ hr = (const v4f*)h;
  float acc = 0.f;
  for (int i = lane; i < D_DIM / 4; i += 32) {
    __builtin_prefetch((const void*)(wr + i + 64), 0, 0);
    v4f w  = __builtin_nontemporal_load(wr + i);
    v4f xv = hr[i];
    acc = fmaf(w.x, xv.x, acc);
    acc = fmaf(w.y, xv.y, acc);
    acc = fmaf(w.z, xv.z, acc);
    acc = fmaf(w.w, xv.w, acc);
  }
#pragma unroll
  for (int off = 16; off; off >>= 1) acc += __shfl_xor(acc, off, 32);
  if (lane == 0) dst[row] = fmaxf(acc + b[row], 0.f);
  if (row == 0 && lane < 7) dst[NF + 4096 + 16384 + lane] = curvs[lane];  // curvs (7)
}

// ---------------------------------------------------------------------------
extern "C" void kernel_launch(void* const* d_in, const int* in_sizes, int n_in,
                              void* d_out, int out_size, void* d_ws, size_t ws_size,
                              hipStream_t stream) {
  (void)in_sizes; (void)n_in; (void)out_size; (void)ws_size;
  const float* x     = (const float*)d_in[0];
  const float* tadj  = (const float*)d_in[1];
  const float* sadj  = (const float*)d_in[2];
  const float* Wc    = (const float*)d_in[3];
  const float* Wt    = (const float*)d_in[4];
  const float* Wtadj = (const float*)d_in[5];
  const float* Ws    = (const float*)d_in[6];
  const float* Wsadj = (const float*)d_in[7];
  const float* Wout  = (const float*)d_in[8];
  const float* Wfcn  = (const float*)d_in[9];
  const float* bfcn  = (const float*)d_in[10];
  const float* curvs = (const float*)d_in[11];
  float* out = (float*)d_out;
  float* ws  = (float*)d_ws;

  float* xc    = ws;                 //  8192 floats
  float* vec   = xc + NF;            // 22016 floats
  float* hvec  = vec + D_DIM;        // 22016 floats
  float* tmp_t = hvec + D_DIM;       //   512 floats
  float* tmp_s = tmp_t + 512;        //  1024 floats   (total ~215 KB)

  // 1) xc = relu(nan_to_num(x) @ Wc^T): 64x128, K=128 -> 4x8 = 32 wave-tiles
  wmma_linear_f32<0, 1><<<4, 256, 0, stream>>>(x, F_DIM, Wc, F_DIM, xc, F_DIM,
                                               /*nTilesX=*/8, /*K=*/F_DIM);
  // 2) adj_t = sigmoid(tadj @ Wtadj^T): 64x64, K=64 -> 16 wave-tiles
  wmma_linear_f32<1, 0><<<2, 256, 0, stream>>>(tadj, N_WIN, Wtadj, N_WIN,
                                               vec + VEC_ADJT, N_WIN, 4, N_WIN);
  // 3) adj_s = sigmoid(sadj @ Wsadj^T): 128x128, K=128 -> 64 wave-tiles
  wmma_linear_f32<1, 0><<<8, 256, 0, stream>>>(sadj, F_DIM, Wsadj, F_DIM,
                                               vec + VEC_ADJS, F_DIM, 8, F_DIM);
  // 4) tiny projections + aggregations -> vec[t_f | s_f]
  proj_small<<<6, 256, 0, stream>>>(xc, Wt, Ws, tmp_t, tmp_s);
  agg_small<<<6, 256, 0, stream>>>(vec, tmp_t, tmp_s);
  // 5) hvec = relu(Wout @ vec)  [1.94 GB stream, HBM-bound] + sigmoid slices
  gemv_out<<<D_DIM / 8, 256, 0, stream>>>(Wout, vec, hvec, out);
  // 6) x_out = relu(Wfcn @ hvec + b)  [0.72 GB stream] + curvs tail
  gemv_fcn<<<NF / 8, 256, 0, stream>>>(Wfcn, hvec, bfcn, curvs, out);
}